// PerformanceModel_4346506903896
// MI455X (gfx1250) — compile-verified
//
#include <hip/hip_runtime.h>
#include <math.h>

#define N_BINS 1024
#define TBL_ELEMS (3 * N_BINS)          // 3072 floats = 12 KB
#define TBL_BYTES (TBL_ELEMS * sizeof(float))
#define EPS 1e-4f
#define BLK 256

// Builtin signature discovered from the round-1 diagnostic:
//   void __builtin_amdgcn_global_load_async_to_lds_b128(v4i AS1*, v4i AS3*, Ii, Ii)
typedef int v4i __attribute__((vector_size(16)));
typedef __attribute__((address_space(1))) v4i as1_v4i;
typedef __attribute__((address_space(3))) v4i as3_v4i;

// ext_vector types: usable with __builtin_nontemporal_load/store (HIP's int4
// is a struct and is rejected by those builtins).
typedef int   ev4i __attribute__((ext_vector_type(4)));
typedef float ev4f __attribute__((ext_vector_type(4)));

// ---------------------------------------------------------------------------
// Kernel 1: build the 3x1024 factor table.
//   f[j][b] = sigmoid((ub[j] - logit(bc[b])) / (ub[j] - lb[j] + EPS))
// One pass over 1024 bins; transcendental cost amortized over 16.7M rows.
// ---------------------------------------------------------------------------
__global__ void build_factor_table(const float* __restrict__ bc,
                                   const float* __restrict__ lb,
                                   const float* __restrict__ ub,
                                   float* __restrict__ tbl) {
  int b = blockIdx.x * blockDim.x + threadIdx.x;
  if (b < N_BINS) {
    float c = bc[b];
    float logit = logf(c / (1.0f - c));
#pragma unroll
    for (int j = 0; j < 3; ++j) {
      float x = (ub[j] - logit) / (ub[j] - lb[j] + EPS);
      tbl[j * N_BINS + b] = 1.0f / (1.0f + expf(-x));
    }
  }
}

// ---------------------------------------------------------------------------
// Shared helper: stage the 12 KB factor table into LDS with CDNA5 async
// global->LDS b128 copies (ASYNCcnt), then barrier.
// ---------------------------------------------------------------------------
__device__ __forceinline__ void stage_table(const float* __restrict__ tbl,
                                            float* lds_tbl) {
#if __has_builtin(__builtin_amdgcn_global_load_async_to_lds_b128)
  {
    // 3072 floats = 768 x 16B chunks; 256 lanes x 3 chunks each.
    int t = threadIdx.x;
#pragma unroll
    for (int k = 0; k < 3; ++k) {
      int off = (k * BLK + t) * 4;  // float index; 16B-aligned chunk start
      __builtin_amdgcn_global_load_async_to_lds_b128(
          (as1_v4i*)(tbl + off),
          (as3_v4i*)(lds_tbl + off),
          0, 0);
    }
#if __has_builtin(__builtin_amdgcn_s_wait_asynccnt)
    __builtin_amdgcn_s_wait_asynccnt(0);
#else
    asm volatile("s_wait_asynccnt 0" ::: "memory");
#endif
  }
  __syncthreads();
#else
  for (int k = threadIdx.x; k < TBL_ELEMS; k += BLK) lds_tbl[k] = tbl[k];
  __syncthreads();
#endif
}

// ---------------------------------------------------------------------------
// Kernel 2: hot loop, 4 rows per iteration.
//   3x nontemporal global_load_b128 (12 indices) -> 12x ds_load_b32 gathers
//   -> 8 v_mul -> 1x nontemporal global_store_b128 (4 results).
// Index/output streams are touch-once: NT cache policy keeps L2 clean.
// ---------------------------------------------------------------------------
__global__ void gather_prod_kernel(const int* __restrict__ idx,
                                   const float* __restrict__ tbl,
                                   float* __restrict__ out,
                                   int n) {
  __shared__ float lds_tbl[TBL_ELEMS];
  stage_table(tbl, lds_tbl);

  const int gid = blockIdx.x * blockDim.x + threadIdx.x;
  const int stride = gridDim.x * blockDim.x;
  const int ngroups = n >> 2;  // groups of 4 rows (48 B of indices each)

  const ev4i* idx4 = (const ev4i*)idx;
  ev4f* out4 = (ev4f*)out;

  for (int g = gid; g < ngroups; g += stride) {
    ev4i a = __builtin_nontemporal_load(&idx4[3 * g + 0]);
    ev4i b = __builtin_nontemporal_load(&idx4[3 * g + 1]);
    ev4i c = __builtin_nontemporal_load(&idx4[3 * g + 2]);
    ev4f r;
    r.x = lds_tbl[a.x] * lds_tbl[N_BINS + a.y] * lds_tbl[2 * N_BINS + a.z];
    r.y = lds_tbl[a.w] * lds_tbl[N_BINS + b.x] * lds_tbl[2 * N_BINS + b.y];
    r.z = lds_tbl[b.z] * lds_tbl[N_BINS + b.w] * lds_tbl[2 * N_BINS + c.x];
    r.w = lds_tbl[c.y] * lds_tbl[N_BINS + c.z] * lds_tbl[2 * N_BINS + c.w];
    __builtin_nontemporal_store(r, &out4[g]);
  }

  // Scalar tail for n % 4 != 0 (dead for n = 16.7M, kept for generality).
  for (int i = (ngroups << 2) + gid; i < n; i += stride) {
    int i0 = idx[3 * i + 0];
    int i1 = idx[3 * i + 1];
    int i2 = idx[3 * i + 2];
    out[i] = lds_tbl[i0] * lds_tbl[N_BINS + i1] * lds_tbl[2 * N_BINS + i2];
  }
}

// ---------------------------------------------------------------------------
// Fallback if workspace is too small for the table: build the table in LDS
// inside each block (12 log/exp per thread, negligible), then same hot loop.
// ---------------------------------------------------------------------------
__global__ void fused_kernel(const float* __restrict__ bc,
                             const int* __restrict__ idx,
                             const float* __restrict__ lb,
                             const float* __restrict__ ub,
                             float* __restrict__ out,
                             int n) {
  __shared__ float lds_tbl[TBL_ELEMS];
  for (int b = threadIdx.x; b < N_BINS; b += blockDim.x) {
    float c = bc[b];
    float logit = logf(c / (1.0f - c));
#pragma unroll
    for (int j = 0; j < 3; ++j) {
      float x = (ub[j] - logit) / (ub[j] - lb[j] + EPS);
      lds_tbl[j * N_BINS + b] = 1.0f / (1.0f + expf(-x));
    }
  }
  __syncthreads();

  const int gid = blockIdx.x * blockDim.x + threadIdx.x;
  const int stride = gridDim.x * blockDim.x;
  for (int i = gid; i < n; i += stride) {
    int i0 = idx[3 * i + 0];
    int i1 = idx[3 * i + 1];
    int i2 = idx[3 * i + 2];
    out[i] = lds_tbl[i0] * lds_tbl[N_BINS + i1] * lds_tbl[2 * N_BINS + i2];
  }
}

extern "C" void kernel_launch(void* const* d_in, const int* in_sizes, int n_in,
                              void* d_out, int out_size, void* d_ws, size_t ws_size,
                              hipStream_t stream) {
  const float* bc  = (const float*)d_in[0];  // bin_centers [1024] f32
  const int*   idx = (const int*)d_in[1];    // indices [n,3] i32
  const float* lb  = (const float*)d_in[2];  // [3] f32
  const float* ub  = (const float*)d_in[3];  // [3] f32
  float* out = (float*)d_out;                // [n] f32
  int n = out_size;

  const int blocks = 4096;  // 1M threads; 4 row-groups (16 rows) per thread

  if (ws_size >= TBL_BYTES) {
    float* tbl = (float*)d_ws;
    build_factor_table<<<(N_BINS + BLK - 1) / BLK, BLK, 0, stream>>>(bc, lb, ub, tbl);
    gather_prod_kernel<<<blocks, BLK, 0, stream>>>(idx, tbl, out, n);
  } else {
    fused_kernel<<<blocks, BLK, 0, stream>>>(bc, idx, lb, ub, out, n);
  }
}